// AvgPoolingMerger_90563680403997
// MI455X (gfx1250) — compile-verified
//
#include <hip/hip_runtime.h>
#include <cstdint>
#include <cstddef>

// Problem constants (match reference).
#define BB 256
#define SS 780
#define DD 1024
#define TT 195   // MAX_TOKENS

// One block per (sample, pooled-token). 256 threads; thread t owns float4 #t
// of the 1024-float output row. Valid tokens stage the 4 source rows into LDS
// via gfx1250 async global->LDS b128 loads (ASYNCcnt), then average.
__global__ __launch_bounds__(256) void AvgPoolingMerger_kernel(
    const float* __restrict__ hs,    // [B,S,D] f32
    const int*   __restrict__ thw,   // [B,3]   i32
    float*       __restrict__ out,   // [B,T,D] f32
    int*         __restrict__ att)   // [B,T]   i32 (aliases tail of d_out)
{
    __shared__ float4 tile[4 * 256];   // 16 KB: 4 gathered rows x 256 float4

    const int j   = blockIdx.x;        // pooled token index
    const int b   = blockIdx.y;        // sample
    const int tid = threadIdx.x;

    // Per-sample pooled-grid geometry (uniform -> SALU).
    const int h  = thw[b * 3 + 1];
    const int w  = thw[b * 3 + 2];
    const int W2 = w >> 1;             // pre-pool grid width
    const int Hp = h >> 2;             // pooled rows   (h/2/2)
    const int Wp = w >> 2;             // pooled cols   (w/2/2)
    const int n_out = Hp * Wp;
    const bool valid = (j < n_out);

    float4* outRow =
        reinterpret_cast<float4*>(out + ((size_t)b * TT + (size_t)j) * DD);

    if (tid == 0) att[b * TT + j] = valid ? 1 : 0;

    if (!valid) {
        // Reference: jnp.where(valid, pooled, 0) -> exact zeros.
        outRow[tid] = make_float4(0.f, 0.f, 0.f, 0.f);
        return;
    }

    // 2x2 window source rows in the (H2,W2) row-major token grid.
    const int r    = j / Wp;
    const int c    = j - r * Wp;
    const int row0 = (2 * r) * W2 + 2 * c;
    const int rows[4] = { row0, row0 + 1, row0 + W2, row0 + W2 + 1 };

    const size_t sampleBase = (size_t)b * SS * DD;

    // Stage 4 x 128-bit per lane into LDS with async loads (no VGPR load
    // destinations held; tracked on ASYNCcnt). Each lane fills only the LDS
    // slots it will read, so no workgroup barrier is required.
#pragma unroll
    for (int k = 0; k < 4; ++k) {
        const float* gp = hs + sampleBase + (size_t)rows[k] * DD + (size_t)tid * 4;
        // LDS flat address: low 32 bits of the generic pointer are the LDS offset.
        uint32_t ldsOff = (uint32_t)(uintptr_t)(&tile[k * 256 + tid]);
        asm volatile("global_load_async_to_lds_b128 %0, %1, off"
                     :
                     : "v"(ldsOff), "v"(gp)
                     : "memory");
    }
    asm volatile("s_wait_asynccnt 0x0" ::: "memory");

    const float4 a = tile[0 * 256 + tid];
    const float4 e = tile[1 * 256 + tid];
    const float4 f = tile[2 * 256 + tid];
    const float4 g = tile[3 * 256 + tid];

    float4 res;
    res.x = (a.x + e.x + f.x + g.x) * 0.25f;
    res.y = (a.y + e.y + f.y + g.y) * 0.25f;
    res.z = (a.z + e.z + f.z + g.z) * 0.25f;
    res.w = (a.w + e.w + f.w + g.w) * 0.25f;

    outRow[tid] = res;   // coalesced global_store_b128
}

extern "C" void kernel_launch(void* const* d_in, const int* in_sizes, int n_in,
                              void* d_out, int out_size, void* d_ws, size_t ws_size,
                              hipStream_t stream) {
    (void)in_sizes; (void)n_in; (void)out_size; (void)d_ws; (void)ws_size;

    const float* hs  = (const float*)d_in[0];   // hidden_states [B,S,D] f32
    // d_in[1] = attention_mask (unused by the math; only its dtype matters)
    const int*   thw = (const int*)d_in[2];     // image_grid_thw [B,3] i32

    float* out = (float*)d_out;                           // [B,T,D] f32
    int*   att = (int*)(out + (size_t)BB * TT * DD);      // [B,T] i32 tail

    dim3 grid(TT, BB);
    dim3 block(256);
    AvgPoolingMerger_kernel<<<grid, block, 0, stream>>>(hs, thw, out, att);
}